// AdExNeuronTorch_65180423684679
// MI455X (gfx1250) — compile-verified
//
#include <hip/hip_runtime.h>
#include <math.h>

// ---------------- problem constants (match reference) ----------------
#define BATCH    16
#define STEPS    2000
#define FEATURES 1024
#define NEURONS  (BATCH * FEATURES)
#define TPB      256
#define NBUF     16            // pipeline depth (STEPS % NBUF == 0)

#define EL_C      (-70.0f)
#define VT_C      (-50.0f)
#define DELTA_T_C (2.0f)
#define INV_DT_SLOPE (0.5f)    // 1/DELTA_T
#define INV_TAU_M (0.05f)      // 1/20
#define INV_TAU_W (0.01f)      // 1/100
#define A_C       (2.0f)
#define V_RESET_C (-58.0f)
#define V_SPIKE_C (20.0f)
#define DT_C      (0.05f)

// address-space-qualified pointer types for the async builtin
typedef __attribute__((address_space(1))) int glob_i32;
typedef __attribute__((address_space(3))) int lds_i32;

// ---------------- wait helpers (guarded; asm fallback) ----------------
__device__ __forceinline__ void wait_async_zero() {
#if defined(__HIP_DEVICE_COMPILE__) && __has_builtin(__builtin_amdgcn_s_wait_asynccnt)
    __builtin_amdgcn_s_wait_asynccnt(0);
#else
    asm volatile("s_wait_asynccnt 0x0" ::: "memory");
#endif
}

__device__ __forceinline__ void wait_ds_zero() {
    asm volatile("s_wait_dscnt 0x0" ::: "memory");
}

// ---------------- one AdEx step (same math/order as reference) --------
__device__ __forceinline__ void adex_step(float I_t, float& v, float& w,
                                          float& v_vis, float& spike) {
    const bool fired = (v >= V_SPIKE_C);
    v_vis = fired ? V_SPIKE_C : v;
    spike = fired ? 1.0f : 0.0f;
    const float v_safe = fired ? V_RESET_C : v;
    const float ex  = DELTA_T_C * expf((v_safe - VT_C) * INV_DT_SLOPE);
    const float dv  = (-(v_safe - EL_C) + ex - w + I_t) * INV_TAU_M;
    const float dw  = (A_C * (v_safe - EL_C) - w) * INV_TAU_W;
    v = fired ? V_RESET_C : v + DT_C * dv;
    w = fired ? w         : w + DT_C * dw;
}

__global__ __launch_bounds__(TPB)
void adex_kernel(const float* __restrict__ I,
                 float* __restrict__ vtr,
                 float* __restrict__ spk) {
    const int n = blockIdx.x * TPB + threadIdx.x;   // 0 .. NEURONS-1
    const int b = n >> 10;                          // n / FEATURES
    const int f = n & (FEATURES - 1);               // n % FEATURES

    const float* gin = I   + (size_t)b * STEPS * FEATURES + f;
    float*       vt  = vtr + (size_t)b * (STEPS + 1) * FEATURES + f;
    float*       sp  = spk + (size_t)b * (STEPS + 1) * FEATURES + f;

    float v = EL_C;     // V_INIT
    float w = 0.0f;     // W_INIT

#if defined(__HIP_DEVICE_COMPILE__) && \
    __has_builtin(__builtin_amdgcn_global_load_async_to_lds_b32)
    // ---------- async-to-LDS double-buffered pipeline (CDNA5 path) ----------
    __shared__ float tile[NBUF][TPB];   // 16 KB; each thread owns one column

    #pragma unroll
    for (int s = 0; s < NBUF; ++s) {
        __builtin_amdgcn_global_load_async_to_lds_b32(
            (glob_i32*)(gin + (size_t)s * FEATURES),
            (lds_i32*)&tile[s][threadIdx.x],
            0, 0);
    }

    for (int t0 = 0; t0 < STEPS; t0 += NBUF) {
        wait_async_zero();                         // current group landed in LDS

        float Ibuf[NBUF];
        #pragma unroll
        for (int s = 0; s < NBUF; ++s) Ibuf[s] = tile[s][threadIdx.x];
        wait_ds_zero();                            // LDS reads done before refill

        const int tn = t0 + NBUF;
        if (tn < STEPS) {
            #pragma unroll
            for (int s = 0; s < NBUF; ++s) {
                __builtin_amdgcn_global_load_async_to_lds_b32(
                    (glob_i32*)(gin + (size_t)(tn + s) * FEATURES),
                    (lds_i32*)&tile[s][threadIdx.x],
                    0, 0);
            }
            // warm L2 two groups ahead (speculative, RT)
            if (tn + NBUF < STEPS)
                __builtin_prefetch(gin + (size_t)(tn + NBUF) * FEATURES, 0, 1);
        }

        #pragma unroll
        for (int s = 0; s < NBUF; ++s) {
            float v_vis, spike;
            adex_step(Ibuf[s], v, w, v_vis, spike);
            const size_t off = (size_t)(t0 + s) * FEATURES;
            __builtin_nontemporal_store(v_vis, vt + off);   // streamed, write-once
            __builtin_nontemporal_store(spike, sp + off);
        }
    }
#else
    // ---------- fallback: register-pipelined non-temporal loads ----------
    float Ibuf[NBUF];
    #pragma unroll
    for (int s = 0; s < NBUF; ++s)
        Ibuf[s] = __builtin_nontemporal_load(gin + (size_t)s * FEATURES);

    for (int t0 = 0; t0 < STEPS; t0 += NBUF) {
        float cur[NBUF];
        #pragma unroll
        for (int s = 0; s < NBUF; ++s) cur[s] = Ibuf[s];

        const int tn = t0 + NBUF;
        if (tn < STEPS) {
            #pragma unroll
            for (int s = 0; s < NBUF; ++s)
                Ibuf[s] = __builtin_nontemporal_load(gin + (size_t)(tn + s) * FEATURES);
            if (tn + NBUF < STEPS)
                __builtin_prefetch(gin + (size_t)(tn + NBUF) * FEATURES, 0, 1);
        }

        #pragma unroll
        for (int s = 0; s < NBUF; ++s) {
            float v_vis, spike;
            adex_step(cur[s], v, w, v_vis, spike);
            const size_t off = (size_t)(t0 + s) * FEATURES;
            __builtin_nontemporal_store(v_vis, vt + off);
            __builtin_nontemporal_store(spike, sp + off);
        }
    }
#endif

    // final slot: raw post-update voltage + its spike flag
    const size_t off = (size_t)STEPS * FEATURES;
    __builtin_nontemporal_store(v, vt + off);
    __builtin_nontemporal_store((v >= V_SPIKE_C) ? 1.0f : 0.0f, sp + off);
}

extern "C" void kernel_launch(void* const* d_in, const int* in_sizes, int n_in,
                              void* d_out, int out_size, void* d_ws, size_t ws_size,
                              hipStream_t stream) {
    (void)in_sizes; (void)n_in; (void)d_ws; (void)ws_size; (void)out_size;
    const float* I = (const float*)d_in[0];
    float* out = (float*)d_out;
    float* vtr = out;                                            // [16,2001,1024]
    float* spk = out + (size_t)BATCH * (STEPS + 1) * FEATURES;   // [16,2001,1024]
    adex_kernel<<<NEURONS / TPB, TPB, 0, stream>>>(I, vtr, spk);
}